// Pooling_11905649345073
// MI455X (gfx1250) — compile-verified
//
#include <hip/hip_runtime.h>
#include <cstdint>
#include <cstddef>

// Problem constants (from reference): B=16, L=4096, D=768, S=128
#define B_  16
#define L_  4096
#define D_  768
#define S_  128
#define SEGCHUNK 16              // segments handled per block in mean kernel
#define FTILES   (D_ / 16)       // 48 feature tiles of 16
#define NCHUNK   (S_ / SEGCHUNK) // 8 chunks per batch
#define FSPLIT   6               // feature-tile groups (8 tiles -> 8 waves each)

typedef float v2f __attribute__((ext_vector_type(2)));
typedef float v8f __attribute__((ext_vector_type(8)));

// ---------------------------------------------------------------------------
// Kernel 0: inclusive cumsum of sent_lengths[b,:] -> offsets[b,:]  (LDS scan)
// ---------------------------------------------------------------------------
__global__ void scan_lengths(const int* __restrict__ lengths,
                             int* __restrict__ offs) {
    __shared__ int buf[S_];
    const int b = blockIdx.x;
    const int t = threadIdx.x;
    buf[t] = lengths[b * S_ + t];
    __syncthreads();
    for (int off = 1; off < S_; off <<= 1) {
        int v = (t >= off) ? buf[t - off] : 0;
        __syncthreads();
        buf[t] += v;
        __syncthreads();
    }
    offs[b * S_ + t] = buf[t];
}

// ---------------------------------------------------------------------------
// Kernel 1: sents_rep gather (+mask) and the concatenated output mask.
// One block per (b,s); 192 threads move 768 floats. The row copy is staged
// through LDS with GLOBAL_LOAD_ASYNC_TO_LDS_B128 (ASYNCcnt-tracked CDNA5
// async path), then masked and stored as float4.
// ---------------------------------------------------------------------------
__global__ void gather_rep(const float* __restrict__ wv,
                           const int* __restrict__ ids,
                           const unsigned char* __restrict__ repMask,
                           const unsigned char* __restrict__ lenMask,
                           float* __restrict__ out,
                           float* __restrict__ outMask) {
    __shared__ float tile[D_];           // 3 KB staging buffer

    const int s = blockIdx.x % S_;
    const int b = blockIdx.x / S_;
    const int idx = ids[b * S_ + s];
    const float m = repMask[b * S_ + s] ? 1.0f : 0.0f;

    const int t = threadIdx.x;           // 0..191 == D/4
    const float* src = wv + ((size_t)b * L_ + (size_t)idx) * D_ + (size_t)t * 4;

#if defined(__HIP_DEVICE_COMPILE__)
    // Per-lane 16B async copy global -> LDS; low 32 bits of the flat shared
    // pointer are the LDS byte offset (ISA 10.2 aperture mapping).
    unsigned ldsOff = (unsigned)(uintptr_t)&tile[t * 4];
    unsigned long long ga = (unsigned long long)(uintptr_t)src;
    asm volatile("global_load_async_to_lds_b128 %0, %1, off"
                 :: "v"(ldsOff), "v"(ga) : "memory");
    asm volatile("s_wait_asynccnt 0x0" ::: "memory");
#else
    tile[t * 4 + 0] = src[0]; tile[t * 4 + 1] = src[1];
    tile[t * 4 + 2] = src[2]; tile[t * 4 + 3] = src[3];
#endif

    float4 v = ((const float4*)tile)[t];   // each lane reads back its own 16B
    v.x *= m; v.y *= m; v.z *= m; v.w *= m;
    float4* dst = (float4*)(out + ((size_t)b * 2 * S_ + (size_t)s) * D_);
    dst[t] = v;

    if (t == 0) outMask[b * 2 * S_ + s] = m;
    if (t == 1) outMask[b * 2 * S_ + S_ + s] = lenMask[b * S_ + s] ? 1.0f : 0.0f;
}

// ---------------------------------------------------------------------------
// Kernel 2: ragged segment mean via V_WMMA_F32_16X16X4_F32.
//   D = A x B + C, A = 16(seg) x 4(token) membership indicator (ragged-exact),
//   B = 4(token) x 16(feature) data slab. Second WMMA chain with B'=(x!=0)
//   accumulates the per-feature nonzero counts.
// Grid: B * NCHUNK * FSPLIT = 768 blocks, 256 threads = 8 waves; each wave
// owns exactly one 16-feature tile (uniform -> EXEC all-ones at the WMMAs).
// Main loop is clamp-free (pointer += stride; second load & prefetch via
// immediate offsets); one uniform clamped tail iteration handles the ragged
// end. Finalize: ballot zero-row test, v_rcp mean, v_cndmask select.
// ---------------------------------------------------------------------------
__global__ void seg_mean_wmma(const float* __restrict__ wv,
                              const int* __restrict__ offs,
                              const unsigned char* __restrict__ lenMask,
                              float* __restrict__ out) {
    __shared__ int   sStart[SEGCHUNK];
    __shared__ int   sEnd[SEGCHUNK];
    __shared__ float sMask[SEGCHUNK];

    const int fblk  = blockIdx.x % FSPLIT;
    const int rest  = blockIdx.x / FSPLIT;
    const int chunk = rest % NCHUNK;
    const int b     = rest / NCHUNK;
    const int s0    = chunk * SEGCHUNK;
    const int tid   = threadIdx.x;

    if (tid < SEGCHUNK) {
        const int s = s0 + tid;
        sStart[tid] = (s == 0) ? 0 : offs[b * S_ + s - 1];
        sEnd[tid]   = offs[b * S_ + s];
        sMask[tid]  = lenMask[b * S_ + s] ? 1.0f : 0.0f;
    }
    __syncthreads();

    const int chunkStart = sStart[0];
    const int chunkEnd   = sEnd[SEGCHUNK - 1];

    const int lane = tid & 31;
    const int wave = tid >> 5;
    const int nm   = lane & 15;   // A: row M; B: col N; D: col N
    const int hi   = lane >> 4;   // selects K pair {0,1} vs {2,3}; D row half

    const float* wvb = wv + (size_t)b * L_ * D_;
    const int ss  = sStart[nm];
    const int se  = sEnd[nm];
    const int ss1 = ss - 1;       // t+1 >= ss  <=>  t >= ss-1
    const int se1 = se - 1;       // t+1 <  se  <=>  t <  se-1

    const int ft = fblk * 8 + wave;        // this wave's feature tile (0..47)
    const int f  = ft * 16 + nm;

    v8f accS = {};   // segment sums   [16 seg x 16 feat]
    v8f accC = {};   // nonzero counts [16 seg x 16 feat]

    // Per-lane streaming pointer for token t0 = tb + 2*hi, feature f.
    const float* p = wvb + (size_t)(chunkStart + hi * 2) * D_ + f;
    int t0 = chunkStart + hi * 2;
    int tb = chunkStart;

    // Main loop: tb+3 < chunkEnd <= L, so no clamping needed.
    for (; tb + 4 <= chunkEnd; tb += 4) {
        const float x0 = p[0];
        const float x1 = p[D_];                 // token t0+1: +3072B imm offset
        __builtin_prefetch(p + 4 * D_, 0, 1);   // next K-group: +12288B

        v2f a, bx, bc;
        a[0]  = (t0 >= ss  && t0 < se ) ? 1.0f : 0.0f;
        a[1]  = (t0 >= ss1 && t0 < se1) ? 1.0f : 0.0f;
        bx[0] = x0;
        bx[1] = x1;
        bc[0] = (x0 != 0.0f) ? 1.0f : 0.0f;
        bc[1] = (x1 != 0.0f) ? 1.0f : 0.0f;

#if defined(__HIP_DEVICE_COMPILE__)
        accS = __builtin_amdgcn_wmma_f32_16x16x4_f32(
            false, a, false, bx, (short)0, accS, false, false);
        accC = __builtin_amdgcn_wmma_f32_16x16x4_f32(
            false, a, false, bc, (short)0, accC, false, false);
#else
        accS[0] += a[0] * bx[0];  // host-pass placeholder only
        accC[0] += a[0] * bc[0];
#endif
        p  += 4 * D_;
        t0 += 4;
    }

    // Tail: 1..3 tokens remain; clamp addresses (indicator A zeroes extras).
    if (tb < chunkEnd) {                        // wave-uniform branch
        const int u0 = min(t0,     L_ - 1);
        const int u1 = min(t0 + 1, L_ - 1);
        const float x0 = wvb[(size_t)u0 * D_ + f];
        const float x1 = wvb[(size_t)u1 * D_ + f];

        v2f a, bx, bc;
        a[0]  = (t0 >= ss  && t0 < se ) ? 1.0f : 0.0f;
        a[1]  = (t0 >= ss1 && t0 < se1) ? 1.0f : 0.0f;
        bx[0] = x0;
        bx[1] = x1;
        bc[0] = (x0 != 0.0f) ? 1.0f : 0.0f;
        bc[1] = (x1 != 0.0f) ? 1.0f : 0.0f;

#if defined(__HIP_DEVICE_COMPILE__)
        accS = __builtin_amdgcn_wmma_f32_16x16x4_f32(
            false, a, false, bx, (short)0, accS, false, false);
        accC = __builtin_amdgcn_wmma_f32_16x16x4_f32(
            false, a, false, bc, (short)0, accC, false, false);
#else
        accS[0] += a[0] * bx[0];
        accC[0] += a[0] * bc[0];
#endif
    }

    // Fallback value word_vectors[0,0,f]: uniform per tile -> hoisted load,
    // so the per-row select below is a pure v_cndmask (no divergence).
    const float fb = wv[f];

    // Finalize: D VGPR r holds row M=r (lanes 0-15) / M=r+8 (lanes 16-31),
    // column N = lane&15. Zero-row test via one wave32 ballot per register.
#pragma unroll
    for (int r = 0; r < 8; ++r) {
        const float sum = accS[r];
        const float cnt = accC[r];

#if defined(__HIP_DEVICE_COMPILE__)
        const unsigned msk = __builtin_amdgcn_ballot_w32(cnt != 0.0f);
        float mean = sum * __builtin_amdgcn_rcpf(fmaxf(cnt, 1.0f));
#else
        const unsigned msk = 1u;                  // host-pass placeholder
        float mean = sum / fmaxf(cnt, 1.0f);
#endif
        const unsigned rowbits = hi ? (msk >> 16) : (msk & 0xFFFFu);
        mean = (rowbits == 0u) ? fb : mean;       // total_nz==0 fallback

        const int m = r + 8 * hi;                 // segment within chunk
        mean *= sMask[m];
        out[((size_t)b * 2 * S_ + S_ + (size_t)(s0 + m)) * D_ + f] = mean;
    }
}

// ---------------------------------------------------------------------------
// Inputs (setup_inputs order):
//   0: word_vectors      f32  [B,L,D]
//   1: sent_rep_token_ids i32 [B,S]
//   2: sent_rep_mask     bool [B,S]
//   3: sent_lengths      i32  [B,S]
//   4: sent_lengths_mask bool [B,S]
// Output: [B,2S,D] f32 followed by [B,2S] mask (as float).
// ---------------------------------------------------------------------------
extern "C" void kernel_launch(void* const* d_in, const int* in_sizes, int n_in,
                              void* d_out, int out_size, void* d_ws, size_t ws_size,
                              hipStream_t stream) {
    (void)in_sizes; (void)n_in; (void)out_size; (void)ws_size;

    const float*         wv      = (const float*)d_in[0];
    const int*           ids     = (const int*)d_in[1];
    const unsigned char* repMask = (const unsigned char*)d_in[2];
    const int*           lengths = (const int*)d_in[3];
    const unsigned char* lenMask = (const unsigned char*)d_in[4];

    float* out     = (float*)d_out;
    float* outMask = out + (size_t)B_ * 2 * S_ * D_;
    int*   offsets = (int*)d_ws;   // B*S ints of scratch

    scan_lengths<<<B_, S_, 0, stream>>>(lengths, offsets);
    gather_rep<<<B_ * S_, D_ / 4, 0, stream>>>(wv, ids, repMask, lenMask, out, outMask);
    seg_mean_wmma<<<B_ * NCHUNK * FSPLIT, 256, 0, stream>>>(wv, offsets, lenMask, out);
}